// DynamicCascadeGNN_30691836297672
// MI455X (gfx1250) — compile-verified
//
#include <hip/hip_runtime.h>

typedef __attribute__((ext_vector_type(2))) float v2f;
typedef __attribute__((ext_vector_type(8))) float v8f;

#define T_ 8
#define N_ 100000
#define E_ 1600000
#define F_ 17
#define G_ 32
#define H_ 48

__device__ __forceinline__ v8f wmma4(v2f a, v2f b, v8f c) {
  // V_WMMA_F32_16X16X4_F32 : D = A(16x4) x B(4x16) + C(16x16), fp32
  return __builtin_amdgcn_wmma_f32_16x16x4_f32(false, a, false, b, (short)0, c,
                                               false, false);
}

__device__ __forceinline__ float sigmoidf_(float x) {
  return 1.0f / (1.0f + __expf(-x));
}

// ---------------------------------------------------------------- zero
__global__ void zero_kernel(float* __restrict__ p, long long n) {
  long long i = (long long)blockIdx.x * blockDim.x + threadIdx.x;
  long long stride = (long long)gridDim.x * blockDim.x;
  for (; i < n; i += stride) p[i] = 0.0f;
}

// ---------------------------------------------------------------- scatter 1
__global__ void scatter1_kernel(const float* __restrict__ x,   // [N,F]
                                const int* __restrict__ src,
                                const int* __restrict__ dst,
                                float* __restrict__ agg,       // [N,F]
                                float* __restrict__ deg) {     // [N]
  int e = blockIdx.x * blockDim.x + threadIdx.x;
  if (e >= E_) return;
  int s = src[e], d = dst[e];
  const float* xs = x + (long long)s * F_;
  float* ad = agg + (long long)d * F_;
#pragma unroll
  for (int f = 0; f < F_; ++f) atomicAdd(&ad[f], xs[f]);
  atomicAdd(&deg[d], 1.0f);
}

// ---------------------------------------------------------------- scatter 2
__global__ void scatter2_kernel(const float* __restrict__ h,   // [N,H]
                                const int* __restrict__ src,
                                const int* __restrict__ dst,
                                float* __restrict__ agg) {     // [N,H]
  int e = blockIdx.x * blockDim.x + threadIdx.x;
  if (e >= E_) return;
  int s = src[e], d = dst[e];
  const float* hs = h + (long long)s * H_;
  float* ad = agg + (long long)d * H_;
#pragma unroll
  for (int f = 0; f < H_; ++f) atomicAdd(&ad[f], hs[f]);
}

// ---------------------------------------------------------------- conv1 (WMMA)
// h1 = relu(x @ Ws^T + bs + (agg/deg) @ Wn^T + bn),  F_=17 -> H_=48
__global__ __launch_bounds__(256) void conv1_kernel(
    const float* __restrict__ x, const float* __restrict__ agg,
    const float* __restrict__ deg, const float* __restrict__ Ws,
    const float* __restrict__ bs, const float* __restrict__ Wn,
    const float* __restrict__ bn, float* __restrict__ out) {
  const int lane = threadIdx.x & 31;
  const int wave = threadIdx.x >> 5;
  const int tile = blockIdx.x * 8 + wave;       // one 16-row tile per wave
  if (tile * 16 >= N_) return;                  // no barriers in this kernel
  const int nb = tile * 16;
  const int m = lane & 15;                      // A-row for this lane
  const int half = lane >> 4;                   // K sub-slot
  const int row = nb + m;
  const int n = lane & 15;                      // output column within tile

  v8f acc0 = {}, acc1 = {}, acc2 = {};
  const float dinv = 1.0f / fmaxf(deg[row], 1.0f);

#pragma unroll
  for (int pass = 0; pass < 2; ++pass) {
    const float* A = pass ? agg : x;
    const float* W = pass ? Wn : Ws;
    const float ascale = pass ? dinv : 1.0f;
    for (int k0 = 0; k0 < F_; k0 += 4) {
      const int kb = k0 + 2 * half;
      v2f a;
      a.x = (kb + 0 < F_) ? A[(long long)row * F_ + kb + 0] * ascale : 0.0f;
      a.y = (kb + 1 < F_) ? A[(long long)row * F_ + kb + 1] * ascale : 0.0f;
      v2f b0, b1, b2;                            // B[k][j] = W[j*F_ + k]
      b0.x = (kb + 0 < F_) ? W[(0 * 16 + n) * F_ + kb + 0] : 0.0f;
      b0.y = (kb + 1 < F_) ? W[(0 * 16 + n) * F_ + kb + 1] : 0.0f;
      b1.x = (kb + 0 < F_) ? W[(1 * 16 + n) * F_ + kb + 0] : 0.0f;
      b1.y = (kb + 1 < F_) ? W[(1 * 16 + n) * F_ + kb + 1] : 0.0f;
      b2.x = (kb + 0 < F_) ? W[(2 * 16 + n) * F_ + kb + 0] : 0.0f;
      b2.y = (kb + 1 < F_) ? W[(2 * 16 + n) * F_ + kb + 1] : 0.0f;
      acc0 = wmma4(a, b0, acc0);
      acc1 = wmma4(a, b1, acc1);
      acc2 = wmma4(a, b2, acc2);
    }
  }
#pragma unroll
  for (int r = 0; r < 8; ++r) {
    const int mr = r + 8 * half;                 // D layout: M = r + 8*half
    long long o = (long long)(nb + mr) * H_;
    out[o + 0 * 16 + n] = fmaxf(acc0[r] + bs[0 * 16 + n] + bn[0 * 16 + n], 0.f);
    out[o + 1 * 16 + n] = fmaxf(acc1[r] + bs[1 * 16 + n] + bn[1 * 16 + n], 0.f);
    out[o + 2 * 16 + n] = fmaxf(acc2[r] + bs[2 * 16 + n] + bn[2 * 16 + n], 0.f);
  }
}

// ------------------------------------------- conv2 + spatial mask (WMMA, fused)
__global__ __launch_bounds__(256) void conv2_mask_kernel(
    const float* __restrict__ h1, const float* __restrict__ agg,
    const float* __restrict__ deg, const float* __restrict__ Ws2,
    const float* __restrict__ bs2, const float* __restrict__ Wn2,
    const float* __restrict__ bn2, const float* __restrict__ Wsm,
    const float* __restrict__ bsm, float* __restrict__ masks_out,
    float* __restrict__ pooled_sum) {
  __shared__ float lds[8][16][H_];               // per-wave 16x48 h2 tile (24KB)
  const int lane = threadIdx.x & 31;
  const int wave = threadIdx.x >> 5;
  const int tile = blockIdx.x * 8 + wave;
  const bool active = (tile * 16 < N_);
  const int nb = active ? tile * 16 : 0;         // inactive waves shadow tile 0
  const int m = lane & 15;
  const int half = lane >> 4;
  const int row = nb + m;
  const int n = lane & 15;

  v8f acc0 = {}, acc1 = {}, acc2 = {};
  const float dinv = 1.0f / fmaxf(deg[row], 1.0f);

#pragma unroll
  for (int pass = 0; pass < 2; ++pass) {
    const float* A = pass ? agg : h1;
    const float* W = pass ? Wn2 : Ws2;
    const float ascale = pass ? dinv : 1.0f;
    for (int k0 = 0; k0 < H_; k0 += 4) {
      const int kb = k0 + 2 * half;
      v2f a;
      a.x = A[(long long)row * H_ + kb + 0] * ascale;
      a.y = A[(long long)row * H_ + kb + 1] * ascale;
      v2f b0, b1, b2;
      b0.x = W[(0 * 16 + n) * H_ + kb + 0]; b0.y = W[(0 * 16 + n) * H_ + kb + 1];
      b1.x = W[(1 * 16 + n) * H_ + kb + 0]; b1.y = W[(1 * 16 + n) * H_ + kb + 1];
      b2.x = W[(2 * 16 + n) * H_ + kb + 0]; b2.y = W[(2 * 16 + n) * H_ + kb + 1];
      acc0 = wmma4(a, b0, acc0);
      acc1 = wmma4(a, b1, acc1);
      acc2 = wmma4(a, b2, acc2);
    }
  }
  // relu(+bias) -> stash h2 tile in LDS (convert D layout -> addressable tile)
#pragma unroll
  for (int r = 0; r < 8; ++r) {
    const int mr = r + 8 * half;
    lds[wave][mr][0 * 16 + n] = fmaxf(acc0[r] + bs2[0 * 16 + n] + bn2[0 * 16 + n], 0.f);
    lds[wave][mr][1 * 16 + n] = fmaxf(acc1[r] + bs2[1 * 16 + n] + bn2[1 * 16 + n], 0.f);
    lds[wave][mr][2 * 16 + n] = fmaxf(acc2[r] + bs2[2 * 16 + n] + bn2[2 * 16 + n], 0.f);
  }
  __syncthreads();

  // second WMMA pass: sm = sigmoid(h2 @ Wsm^T + bsm)
  v8f s0 = {}, s1 = {}, s2 = {};
  for (int k0 = 0; k0 < H_; k0 += 4) {
    const int kb = k0 + 2 * half;
    v2f a;
    a.x = lds[wave][m][kb + 0];
    a.y = lds[wave][m][kb + 1];
    v2f b0, b1, b2;
    b0.x = Wsm[(0 * 16 + n) * H_ + kb + 0]; b0.y = Wsm[(0 * 16 + n) * H_ + kb + 1];
    b1.x = Wsm[(1 * 16 + n) * H_ + kb + 0]; b1.y = Wsm[(1 * 16 + n) * H_ + kb + 1];
    b2.x = Wsm[(2 * 16 + n) * H_ + kb + 0]; b2.y = Wsm[(2 * 16 + n) * H_ + kb + 1];
    s0 = wmma4(a, b0, s0);
    s1 = wmma4(a, b1, s1);
    s2 = wmma4(a, b2, s2);
  }

  float psum0 = 0.f, psum1 = 0.f, psum2 = 0.f;
#pragma unroll
  for (int r = 0; r < 8; ++r) {
    const int mr = r + 8 * half;
    const float sm0 = sigmoidf_(s0[r] + bsm[0 * 16 + n]);
    const float sm1 = sigmoidf_(s1[r] + bsm[1 * 16 + n]);
    const float sm2 = sigmoidf_(s2[r] + bsm[2 * 16 + n]);
    psum0 += lds[wave][mr][0 * 16 + n] * sm0;
    psum1 += lds[wave][mr][1 * 16 + n] * sm1;
    psum2 += lds[wave][mr][2 * 16 + n] * sm2;
    if (active) {
      long long o = (long long)(nb + mr) * H_;
      masks_out[o + 0 * 16 + n] = sm0;
      masks_out[o + 1 * 16 + n] = sm1;
      masks_out[o + 2 * 16 + n] = sm2;
    }
  }
  // combine the two row-halves (M 0..7 vs 8..15) per column, then accumulate
  psum0 += __shfl_xor(psum0, 16, 32);
  psum1 += __shfl_xor(psum1, 16, 32);
  psum2 += __shfl_xor(psum2, 16, 32);
  if (active && half == 0) {
    atomicAdd(&pooled_sum[0 * 16 + n], psum0);
    atomicAdd(&pooled_sum[1 * 16 + n], psum1);
    atomicAdd(&pooled_sum[2 * 16 + n], psum2);
  }
}

// ---------------------------------------------------------------- head (tiny)
__global__ __launch_bounds__(64) void head_kernel(
    const float* __restrict__ gf, const float* __restrict__ Wgp,
    const float* __restrict__ bgp, const float* __restrict__ W_ih,
    const float* __restrict__ b_ih, const float* __restrict__ W_hh,
    const float* __restrict__ b_hh, const float* __restrict__ Wa,
    const float* __restrict__ ba, const float* __restrict__ Wtm,
    const float* __restrict__ btm, const float* __restrict__ Wr1,
    const float* __restrict__ br1, const float* __restrict__ Wr2,
    const float* __restrict__ br2, const float* __restrict__ Ws1,
    const float* __restrict__ pooled_sum, float* __restrict__ out) {
  __shared__ float seq[T_][2 * H_];
  __shared__ float hcur[H_];
  __shared__ float gout[T_][H_];
  __shared__ float scores[T_], wts[T_], tmask[T_];
  __shared__ float context[H_], r1[H_];
  const int j = threadIdx.x;

  if (j < H_) {
    for (int t = 0; t < T_; ++t) {
      seq[t][j] = pooled_sum[t * H_ + j] / (float)N_;
      float g = bgp[j];
      for (int k = 0; k < G_; ++k) g += gf[t * G_ + k] * Wgp[j * G_ + k];
      seq[t][H_ + j] = fmaxf(g, 0.f);
    }
    hcur[j] = 0.f;
  }
  __syncthreads();

  for (int t = 0; t < T_; ++t) {
    float hnew = 0.f;
    if (j < H_) {
      float gir = b_ih[j], giz = b_ih[H_ + j], gin = b_ih[2 * H_ + j];
      for (int k = 0; k < 2 * H_; ++k) {
        const float xv = seq[t][k];
        gir += W_ih[(0 * H_ + j) * (2 * H_) + k] * xv;
        giz += W_ih[(1 * H_ + j) * (2 * H_) + k] * xv;
        gin += W_ih[(2 * H_ + j) * (2 * H_) + k] * xv;
      }
      float ghr = b_hh[j], ghz = b_hh[H_ + j], ghn = b_hh[2 * H_ + j];
      for (int k = 0; k < H_; ++k) {
        const float hv = hcur[k];
        ghr += W_hh[(0 * H_ + j) * H_ + k] * hv;
        ghz += W_hh[(1 * H_ + j) * H_ + k] * hv;
        ghn += W_hh[(2 * H_ + j) * H_ + k] * hv;
      }
      const float r = 1.f / (1.f + expf(-(gir + ghr)));
      const float z = 1.f / (1.f + expf(-(giz + ghz)));
      const float nn = tanhf(gin + r * ghn);
      hnew = (1.f - z) * nn + z * hcur[j];
    }
    __syncthreads();
    if (j < H_) { hcur[j] = hnew; gout[t][j] = hnew; }
    __syncthreads();
  }

  if (j < T_) {
    float sc = ba[0], tm = btm[0];
    for (int k = 0; k < H_; ++k) {
      sc += gout[j][k] * Wa[k];
      tm += gout[j][k] * Wtm[k];
    }
    scores[j] = sc;
    tmask[j] = 1.f / (1.f + expf(-tm));
  }
  __syncthreads();
  if (j == 0) {
    float mx = scores[0];
    for (int t = 1; t < T_; ++t) mx = fmaxf(mx, scores[t]);
    float s = 0.f;
    for (int t = 0; t < T_; ++t) { wts[t] = expf(scores[t] - mx); s += wts[t]; }
    for (int t = 0; t < T_; ++t) wts[t] /= s;
  }
  __syncthreads();
  if (j < H_) {
    float c = 0.f;
    for (int t = 0; t < T_; ++t) c += gout[t][j] * wts[t] * tmask[t];
    context[j] = c;
  }
  __syncthreads();
  if (j < H_) {
    float v = br1[j];
    for (int k = 0; k < H_; ++k) v += Wr1[j * H_ + k] * context[k];
    r1[j] = fmaxf(v, 0.f);
  }
  __syncthreads();
  if (j == 0) {
    float p = br2[0];
    for (int k = 0; k < H_; ++k) p += r1[k] * Wr2[k];
    out[0] = p;                                        // pred
    for (int t = 0; t < T_; ++t) out[1 + t] = wts[t];  // weights
    // top8 of mean(|Ws1|, axis=0) over F_=17 channels, descending
    float cm[F_];
    for (int f = 0; f < F_; ++f) {
      float s = 0.f;
      for (int h = 0; h < H_; ++h) s += fabsf(Ws1[h * F_ + f]);
      cm[f] = s / (float)H_;
    }
    for (int i = 0; i < 8; ++i) {
      int bi = 0; float bv = -1e30f;
      for (int f = 0; f < F_; ++f)
        if (cm[f] > bv) { bv = cm[f]; bi = f; }
      out[9 + i] = bv;
      cm[bi] = -1e31f;
    }
    const long long tmoff = 17LL + (long long)T_ * N_ * H_;
    for (int t = 0; t < T_; ++t) out[tmoff + t] = tmask[t];  // temp_mask
  }
}

// ---------------------------------------------------------------- launch
extern "C" void kernel_launch(void* const* d_in, const int* in_sizes, int n_in,
                              void* d_out, int out_size, void* d_ws,
                              size_t ws_size, hipStream_t stream) {
  (void)in_sizes; (void)n_in; (void)out_size; (void)ws_size;
  const float* nf  = (const float*)d_in[0];
  const int*   ei  = (const int*)d_in[1];
  const float* gf  = (const float*)d_in[2];
  const float* Ws1 = (const float*)d_in[3];  const float* bs1 = (const float*)d_in[4];
  const float* Wn1 = (const float*)d_in[5];  const float* bn1 = (const float*)d_in[6];
  const float* Ws2 = (const float*)d_in[7];  const float* bs2 = (const float*)d_in[8];
  const float* Wn2 = (const float*)d_in[9];  const float* bn2 = (const float*)d_in[10];
  const float* Wgp = (const float*)d_in[11]; const float* bgp = (const float*)d_in[12];
  const float* W_ih= (const float*)d_in[13]; const float* b_ih= (const float*)d_in[14];
  const float* W_hh= (const float*)d_in[15]; const float* b_hh= (const float*)d_in[16];
  const float* Wa  = (const float*)d_in[17]; const float* ba  = (const float*)d_in[18];
  const float* Wsm = (const float*)d_in[19]; const float* bsm = (const float*)d_in[20];
  const float* Wtm = (const float*)d_in[21]; const float* btm = (const float*)d_in[22];
  const float* Wr1 = (const float*)d_in[23]; const float* br1 = (const float*)d_in[24];
  const float* Wr2 = (const float*)d_in[25]; const float* br2 = (const float*)d_in[26];

  float* out = (float*)d_out;
  float* ws  = (float*)d_ws;
  float* agg1   = ws;                                   // N*F
  float* deg    = agg1 + (size_t)N_ * F_;               // N
  float* agg2   = deg + N_;                             // N*H
  float* h1     = agg2 + (size_t)N_ * H_;               // N*H
  float* pooled = h1 + (size_t)N_ * H_;                 // T*H

  zero_kernel<<<1, 256, 0, stream>>>(pooled, (long long)T_ * H_);

  const int eblocks = (E_ + 255) / 256;
  const int cblocks = ((N_ / 16) + 7) / 8;              // 8 waves/block, 16 rows/wave

  for (int t = 0; t < T_; ++t) {
    const float* xt  = nf + (size_t)t * N_ * F_;
    const int* srct  = ei + (size_t)t * 2 * E_;
    const int* dstt  = srct + E_;
    zero_kernel<<<2048, 256, 0, stream>>>(agg1, (long long)N_ * F_ + N_);
    scatter1_kernel<<<eblocks, 256, 0, stream>>>(xt, srct, dstt, agg1, deg);
    conv1_kernel<<<cblocks, 256, 0, stream>>>(xt, agg1, deg, Ws1, bs1, Wn1, bn1, h1);
    zero_kernel<<<2048, 256, 0, stream>>>(agg2, (long long)N_ * H_);
    scatter2_kernel<<<eblocks, 256, 0, stream>>>(h1, srct, dstt, agg2);
    conv2_mask_kernel<<<cblocks, 256, 0, stream>>>(
        h1, agg2, deg, Ws2, bs2, Wn2, bn2, Wsm, bsm,
        out + 17 + (size_t)t * N_ * H_, pooled + (size_t)t * H_);
  }

  head_kernel<<<1, 64, 0, stream>>>(gf, Wgp, bgp, W_ih, b_ih, W_hh, b_hh, Wa, ba,
                                    Wtm, btm, Wr1, br1, Wr2, br2, Ws1, pooled, out);
}